// patch_expanding_66881230733751
// MI455X (gfx1250) — compile-verified
//
#include <hip/hip_runtime.h>
#include <stdint.h>

typedef float v2f __attribute__((ext_vector_type(2)));
typedef float v8f __attribute__((ext_vector_type(8)));

// Problem: H=32, W=32, C=1024, UP=2, B=32
//   GEMM: Y[65536, 512] = Xi[65536, 512] * W[512, 512]
//   Xi row r = b*2048 + h*64 + u*32 + wc maps to x row b*1024 + h*32 + wc,
//   channel offset u*512. Each aligned 16-row tile -> 16 contiguous x rows.
#define KTOT 512
#define NTOT 512
#define KCHUNK 64
// LDS layout: k-pair-interleaved W chunk. Element (k,n) of the chunk lives at
// wlds[(k>>1)*PSTRIDE + 2*n + (k&1)]. PSTRIDE = 1056 words: 1024 data + 32
// pad; 1056 % 64 == 32 so pair rows p and p+1 occupy disjoint 32-bank
// windows -> conflict-free ds_load_b64 across the two half-waves.
#define PSTRIDE 1056
#define LDS_FLOATS ((KCHUNK / 2) * PSTRIDE)  // 33792 floats = 132 KB

__global__ __launch_bounds__(256) void patch_expanding_gemm_kernel(
    const float* __restrict__ x, const float* __restrict__ w,
    float* __restrict__ out) {
  extern __shared__ float wlds[];

  const int tid    = threadIdx.x;
  const int wave   = tid >> 5;
  const int lane   = tid & 31;
  const int lanelo = lane & 15;
  const int ksel   = lane >> 4;  // 0: lanes 0-15, 1: lanes 16-31

  const int wm = wave & 1;   // M sub-tile within block
  const int wn = wave >> 1;  // N sub-tile within block (0..3)
  const int m0 = blockIdx.x * 32 + wm * 16;  // output row tile base
  const int n0 = wn * 128;                   // output col tile base

  // Decompose output row -> input row of x (channel-split interleave)
  const int b  = m0 >> 11;
  const int h  = (m0 >> 6) & 31;
  const int u  = (m0 >> 5) & 1;
  const int wc = (m0 & 31) + lanelo;

  // A fragment (16x4 f32): lane holds row M=lanelo, K = k + 2*ksel + {0,1}
  const float* aptr =
      x + (size_t)((b * 32 + h) * 32 + wc) * 1024 + u * 512 + 2 * ksel;

  // Low 32 bits of a generic LDS pointer == LDS byte offset (shared aperture
  // base lives in addr[63:32]).
  const unsigned lds_base = (unsigned)(uintptr_t)wlds;

  v8f acc[8] = {};

  for (int kc = 0; kc < KTOT; kc += KCHUNK) {
    __syncthreads();  // previous chunk fully consumed

    // Stage W[kc..kc+64) x [0..512) into LDS, pair-interleaved, using
    // async copies (ASYNCcnt): coalesced global b32 reads, per-lane LDS
    // scatter destinations. 64*512 elements / 256 threads = 128 each.
    {
      const unsigned long long wbase =
          (unsigned long long)(uintptr_t)(w + (size_t)kc * NTOT);
#pragma unroll 4
      for (int i = 0; i < 128; ++i) {
        int e  = i * 256 + tid;   // linear element in row-major chunk
        int kr = e >> 9;          // chunk row 0..63
        int c  = e & 511;         // column
        unsigned ldsa =
            lds_base + (unsigned)(((kr >> 1) * PSTRIDE + 2 * c + (kr & 1)) * 4);
        unsigned voff = (unsigned)(e * 4);  // byte offset from wbase
        asm volatile("global_load_async_to_lds_b32 %0, %1, %2"
                     :
                     : "v"(ldsa), "v"(voff), "s"(wbase)
                     : "memory");
      }
    }
    asm volatile("s_wait_asynccnt 0x0" ::: "memory");
    __syncthreads();  // chunk visible to all waves

#pragma unroll 4
    for (int kl = 0; kl < KCHUNK; kl += 4) {
      v2f a = *(const v2f*)(aptr + kc + kl);  // global b64, A 16x4 fragment
      // B fragment: one contiguous b64 per sub-tile; pair row p = kl/2 + ksel
      const float* wl = wlds + ((kl >> 1) + ksel) * PSTRIDE + 2 * lanelo;
#pragma unroll
      for (int j = 0; j < 8; ++j) {
        v2f bv = *(const v2f*)(wl + 2 * (n0 + j * 16));
        acc[j] = __builtin_amdgcn_wmma_f32_16x16x4_f32(
            false, a, false, bv, (short)0, acc[j], false, false);
      }
    }
  }

  // C/D layout: VGPR v holds row m0 + v + 8*ksel, lane holds col n0+j*16+lanelo
#pragma unroll
  for (int j = 0; j < 8; ++j) {
    float* op = out + (size_t)(m0 + 8 * ksel) * NTOT + n0 + j * 16 + lanelo;
#pragma unroll
    for (int v = 0; v < 8; ++v) {
      op[(size_t)v * NTOT] = acc[j][v];
    }
  }
}

extern "C" void kernel_launch(void* const* d_in, const int* in_sizes, int n_in,
                              void* d_out, int out_size, void* d_ws,
                              size_t ws_size, hipStream_t stream) {
  const float* x = (const float*)d_in[0];  // (32, 1024, 1024) f32
  const float* w = (const float*)d_in[1];  // (512, 512) f32
  float* out = (float*)d_out;              // (32, 2048, 512) f32

  dim3 grid(65536 / 32);  // 2048 blocks: 32 rows x all 512 cols each
  dim3 block(256);
  size_t lds_bytes = (size_t)LDS_FLOATS * sizeof(float);  // 132 KB
  patch_expanding_gemm_kernel<<<grid, block, lds_bytes, stream>>>(x, w, out);
}